// MethodDCT_16166256902885
// MI455X (gfx1250) — compile-verified
//
#include <hip/hip_runtime.h>
#include <stdint.h>

typedef __attribute__((ext_vector_type(2))) float v2f;
typedef __attribute__((ext_vector_type(8))) float v8f;

namespace {

constexpr int   kN          = 512;
constexpr float kPiOver1024 = 3.14159265358979323846f / 1024.0f;
// Both DCT matrices carry a factor 2; fold 2*2 = 4 into the 1/1000 scale.
constexpr float kScale      = 4.0e-3f;

// LDS layout (floats): [0, 32768)      tmp   64x512 intermediate
//                      [32768, 40960)  per-wave double-buffered X stripes
//                      total 160 KB -> 2 workgroups per 320 KB WGP
constexpr int kTmpFloats     = 64 * kN;           // 32768
constexpr int kStagePerWave  = 1024;              // 2 buffers x 512 floats
constexpr int kStageBufBytes = 2048;              // 8 rows x 64 cols x 4B

// cos(pi*(2n+1)*k/1024) without the factor 2. The angle has exact period
// 2048 in t = (2n+1)*k; v_cos_f32 computes cos(2*pi*x), so t * 2^-11 is an
// EXACT range reduction into [0,1) — one VALU op and better accuracy than
// a large-angle cosf.
__device__ __forceinline__ float dct_cos(int k, int n) {
    const int t = ((2 * n + 1) * k) & 2047;
#if __has_builtin(__builtin_amdgcn_cosf)
    return __builtin_amdgcn_cosf((float)t * 0x1p-11f);
#else
    return __cosf((float)t * kPiOver1024);
#endif
}

#if __has_builtin(__builtin_amdgcn_s_wait_asynccnt)
#define WAIT_ASYNCCNT(n) __builtin_amdgcn_s_wait_asynccnt(n)
#else
#define WAIT_ASYNCCNT(n) asm volatile("s_wait_asynccnt %0" ::"i"(n) : "memory")
#endif

// Async-copy one 8-row x 64-col stripe of X (2 KB) into this wave's LDS
// staging buffer: 128 x 16B chunks, 4 per lane, fully coalesced.
// Tracked with ASYNCcnt (4 increments per call).
__device__ __forceinline__ void stage_issue(const float* __restrict__ X,
                                            int kb, int colBase, int lane,
                                            unsigned stageByteBase, int buf) {
#pragma unroll
    for (int j = 0; j < 4; ++j) {
        const int c   = lane + 32 * j;   // chunk id 0..127
        const int row = c >> 4;          // 0..7
        const int cc  = c & 15;          // 16B chunk within the 64-col row
        const float* g =
            X + (size_t)(kb * 8 + row) * kN + colBase + cc * 4;
        const unsigned ldsByte =
            stageByteBase + (unsigned)buf * kStageBufBytes + (unsigned)c * 16;
        asm volatile("global_load_async_to_lds_b128 %0, %1, off"
                     ::"v"(ldsByte),
                       "v"((unsigned long long)(uintptr_t)g)
                     : "memory");
    }
}

} // namespace

// One workgroup = one (b,c) image x one 64-row output block.
// 256 threads = 8 waves; each wave owns a 64-column stripe of the output.
// Stage 1: tmp = C_rows(64x512) @ X   (A = on-the-fly cosines,
//                                      B = async-staged LDS stripe of X)
// Stage 2: out = tmp @ C^T * 4e-3     (A = ds_load_b64 from tmp,
//                                      B = on-the-fly cosines)
__global__ __launch_bounds__(256)
void dct2d_wmma_f32_kernel(const float* __restrict__ img,
                           float* __restrict__ out) {
    extern __shared__ float lds[];

    const int wg   = blockIdx.x;
    const int ic   = wg >> 3;               // flattened (b,c) image index
    const int rb   = wg & 7;                // 64-row block within the image
    const int row0 = rb * 64;

    const float* __restrict__ X = img + (size_t)ic * kN * kN;
    float* __restrict__       O = out + (size_t)ic * kN * kN;

    const int tid     = threadIdx.x;
    const int wave    = tid >> 5;
    const int lane    = tid & 31;
    const int laneM   = lane & 15;          // row (A) / column (B,C,D) in tile
    const int laneK   = (lane >> 4) * 2;    // K sub-offset per lane half
    const int rOff    = (lane >> 4) * 8;    // C/D row offset for upper half
    const int colBase = wave * 64;          // this wave's first output column

    float* tmp   = lds;                                   // 64 x 512
    float* stage = lds + kTmpFloats + wave * kStagePerWave;
    const unsigned stageByteBase =
        (unsigned)(kTmpFloats + wave * kStagePerWave) * sizeof(float);

    const v8f vzero = {0.f, 0.f, 0.f, 0.f, 0.f, 0.f, 0.f, 0.f};
    v8f acc[4][4];                          // [ntile][mtile] 16x16 f32 tiles

    // ---------------- Stage 1: tmp = C_block(64x512) @ X(512x512) --------
#pragma unroll
    for (int nt = 0; nt < 4; ++nt)
#pragma unroll
        for (int mt = 0; mt < 4; ++mt)
            acc[nt][mt] = vzero;

    stage_issue(X, 0, colBase, lane, stageByteBase, 0);   // prologue

    for (int kb = 0; kb < kN / 8; ++kb) {                 // 64 k-blocks
        const int buf = kb & 1;
        const bool more = (kb + 1) < (kN / 8);
        if (more)
            stage_issue(X, kb + 1, colBase, lane, stageByteBase, buf ^ 1);

        // L2 warming two blocks ahead (global_prefetch_b8).
        if (kb + 2 < kN / 8) {
            const float* pf = X + (size_t)((kb + 2) * 8 + (lane >> 2)) * kN +
                              colBase + (lane & 3) * 16;
            __builtin_prefetch(pf, 0, 1);
        }

        // Wave-local completion: keep next block's 4 copies in flight.
        if (more) { WAIT_ASYNCCNT(4); } else { WAIT_ASYNCCNT(0); }

        const float* sb = stage + buf * (kStageBufBytes / 4);
#pragma unroll
        for (int k0l = 0; k0l < 8; k0l += 4) {
            const int h  = kb * 8 + k0l + laneK;  // global K index
            const int hl = k0l + laneK;           // K index within buffer

            // A fragments: DCT basis rows for this 64-row block.
            v2f a[4];
#pragma unroll
            for (int mt = 0; mt < 4; ++mt) {
                const int krow = row0 + mt * 16 + laneM;
                a[mt].x = dct_cos(krow, h);
                a[mt].y = dct_cos(krow, h + 1);
            }

#pragma unroll
            for (int nt = 0; nt < 4; ++nt) {
                const int cidx = nt * 16 + laneM;
                v2f b;                           // B fragment from LDS stripe
                b.x = sb[hl * 64 + cidx];
                b.y = sb[(hl + 1) * 64 + cidx];
#pragma unroll
                for (int mt = 0; mt < 4; ++mt) {
                    acc[nt][mt] = __builtin_amdgcn_wmma_f32_16x16x4_f32(
                        false, a[mt], false, b, (short)0, acc[nt][mt], false,
                        false);
                }
            }
        }
    }

    // Spill stage-1 tiles into LDS tmp (each wave owns disjoint columns).
#pragma unroll
    for (int nt = 0; nt < 4; ++nt) {
        const int col = colBase + nt * 16 + laneM;
#pragma unroll
        for (int mt = 0; mt < 4; ++mt) {
#pragma unroll
            for (int v = 0; v < 8; ++v) {
                tmp[(mt * 16 + rOff + v) * kN + col] = acc[nt][mt][v];
            }
        }
    }
    __syncthreads();

    // ---------------- Stage 2: out = tmp(64x512) @ C^T, scaled -----------
#pragma unroll
    for (int nt = 0; nt < 4; ++nt)
#pragma unroll
        for (int mt = 0; mt < 4; ++mt)
            acc[nt][mt] = vzero;

    for (int k0 = 0; k0 < kN; k0 += 4) {
        const int w = k0 + laneK;

        // A fragments from tmp: 8-byte aligned v2f loads (ds_load_b64).
        v2f a[4];
#pragma unroll
        for (int mt = 0; mt < 4; ++mt) {
            a[mt] = *(const v2f*)&tmp[(mt * 16 + laneM) * kN + w];
        }

#pragma unroll
        for (int nt = 0; nt < 4; ++nt) {
            const int l = colBase + nt * 16 + laneM;  // output column
            v2f b;                                    // B[w][l] = C[l][w]
            b.x = dct_cos(l, w);
            b.y = dct_cos(l, w + 1);
#pragma unroll
            for (int mt = 0; mt < 4; ++mt) {
                acc[nt][mt] = __builtin_amdgcn_wmma_f32_16x16x4_f32(
                    false, a[mt], false, b, (short)0, acc[nt][mt], false,
                    false);
            }
        }
    }

    // Store scaled result (every output element covered exactly once).
#pragma unroll
    for (int nt = 0; nt < 4; ++nt) {
        const int col = colBase + nt * 16 + laneM;
#pragma unroll
        for (int mt = 0; mt < 4; ++mt) {
#pragma unroll
            for (int v = 0; v < 8; ++v) {
                const int row = row0 + mt * 16 + rOff + v;
                O[(size_t)row * kN + col] = acc[nt][mt][v] * kScale;
            }
        }
    }
}

extern "C" void kernel_launch(void* const* d_in, const int* in_sizes, int n_in,
                              void* d_out, int out_size, void* d_ws, size_t ws_size,
                              hipStream_t stream) {
    (void)n_in; (void)out_size; (void)d_ws; (void)ws_size;

    const float* img = (const float*)d_in[0];
    float*       out = (float*)d_out;

    const int n_images = in_sizes[0] / (kN * kN);   // 32*3 = 96
    const dim3 grid(n_images * 8);                  // 8 row-blocks per image
    const dim3 block(256);
    const size_t lds_bytes =
        (size_t)(kTmpFloats + 8 * kStagePerWave) * sizeof(float);  // 160 KB

    // Allow >64KB dynamic LDS (host-side attribute; not a stream op, so
    // graph-capture safe and deterministic on every call).
    (void)hipFuncSetAttribute((const void*)dct2d_wmma_f32_kernel,
                              hipFuncAttributeMaxDynamicSharedMemorySize,
                              (int)lds_bytes);

    dct2d_wmma_f32_kernel<<<grid, block, lds_bytes, stream>>>(img, out);
}